// VarRnnARPredictor_67783173865530
// MI455X (gfx1250) — compile-verified
//
#include <hip/hip_runtime.h>
#include <hip/hip_bf16.h>
#include <stdint.h>

// ---------------------------------------------------------------------------
// VarRnnARPredictor for MI455X (gfx1250, wave32, WMMA)
// B=32, T=2000, C=256, P=256, H=512, 4H=2048, D0=P+C=512
// ---------------------------------------------------------------------------

typedef float v2f __attribute__((ext_vector_type(2)));
typedef float v8f __attribute__((ext_vector_type(8)));

#define Bn    32
#define Tn    2000
#define Cn    256
#define Pn    256
#define Hn    512
#define Gn    2048           // 4*H
#define D0n   512            // P + C
#define Mrows (Bn * Tn)      // 64000
#define NBLK_REC 32          // persistent blocks in recurrent kernel

// fp32 WMMA: D(16x16) = A(16x4) * B(4x16) + C, full fp32 accuracy.
__device__ __forceinline__ v8f wmma4(v2f a, v2f b, v8f c) {
    return __builtin_amdgcn_wmma_f32_16x16x4_f32(
        /*neg_a=*/false, a, /*neg_b=*/false, b,
        /*c_mod=*/(short)0, c, /*reuse_a=*/false, /*reuse_b=*/false);
}

__device__ __forceinline__ float sigmoidf_fast(float x) {
    return 1.0f / (1.0f + __expf(-x));
}

// ---------------------------------------------------------------------------
// init: zero step counters and ping-pong h buffers
// ---------------------------------------------------------------------------
__global__ void k_init(int* __restrict__ cnt, float* __restrict__ hb) {
    int i = blockIdx.x * blockDim.x + threadIdx.x;
    if (i < Tn) cnt[i] = 0;
    for (int j = i; j < 2 * Bn * Hn; j += gridDim.x * blockDim.x) hb[j] = 0.0f;
}

// ---------------------------------------------------------------------------
// prenet: x0[:, 0:256] = relu(relu(inp*pw0+pb0) @ pw1.T + pb1)
//   A is generated on the fly (input dim of layer0 is 1).
// grid: (Mrows/128, Pn/16), block 256 (8 waves; wave w -> m-tile w)
// ---------------------------------------------------------------------------
__global__ __launch_bounds__(256)
void k_prenet(const float* __restrict__ inp,  const float* __restrict__ pw0,
              const float* __restrict__ pb0,  const float* __restrict__ pw1,
              const float* __restrict__ pb1,  float* __restrict__ x0) {
    __shared__ float sW[16 * (Pn + 4)];   // pw1 tile, padded for banks
    __shared__ float sw0[Pn];
    __shared__ float sb0[Pn];

    const int tid   = threadIdx.x;
    const int n0    = blockIdx.y * 16;
    const int mBase = blockIdx.x * 128;

    for (int t4 = tid; t4 < 16 * (Pn / 4); t4 += 256) {
        int nl = t4 / (Pn / 4), kq = t4 % (Pn / 4);
        float4 v = *(const float4*)(pw1 + (size_t)(n0 + nl) * Pn + kq * 4);
        *(float4*)(&sW[nl * (Pn + 4) + kq * 4]) = v;
    }
    if (tid < Pn) { sw0[tid] = pw0[tid]; sb0[tid] = pb0[tid]; }
    __syncthreads();

    const int wave  = tid >> 5, lane = tid & 31;
    const int mrow  = lane & 15;
    const int khalf = lane >> 4;          // lanes 16-31 hold K+2
    const int ncol  = lane & 15;
    const int mW    = mBase + wave * 16;
    const float am  = inp[mW + mrow];     // scalar input for this row
    const float* bRow = &sW[ncol * (Pn + 4) + 2 * khalf];

    v8f a0 = {}, a1 = {};
    #pragma unroll 4
    for (int k0 = 0; k0 < Pn; k0 += 8) {
        int k = k0 + 2 * khalf;
        v2f a; v2f b;
        a.x = fmaxf(fmaf(am, sw0[k],     sb0[k]),     0.0f);
        a.y = fmaxf(fmaf(am, sw0[k + 1], sb0[k + 1]), 0.0f);
        b = *(const v2f*)(bRow + k0);
        a0 = wmma4(a, b, a0);
        k = k0 + 4 + 2 * khalf;
        a.x = fmaxf(fmaf(am, sw0[k],     sb0[k]),     0.0f);
        a.y = fmaxf(fmaf(am, sw0[k + 1], sb0[k + 1]), 0.0f);
        b = *(const v2f*)(bRow + k0 + 4);
        a1 = wmma4(a, b, a1);
    }
    v8f acc = a0 + a1;

    const int n = n0 + ncol;
    const float bias = pb1[n];
    #pragma unroll
    for (int r = 0; r < 8; ++r) {
        int mm = mW + r + 8 * khalf;
        x0[(size_t)mm * D0n + n] = fmaxf(acc[r] + bias, 0.0f);
    }
}

// ---------------------------------------------------------------------------
// copy cond into x0[:, 256:512]
// ---------------------------------------------------------------------------
__global__ void k_condcopy(const float* __restrict__ cond, float* __restrict__ x0) {
    int i = blockIdx.x * blockDim.x + threadIdx.x;
    const int total = Mrows * (Cn / 4);
    if (i < total) {
        int row = i / (Cn / 4), c4 = i % (Cn / 4);
        float4 v = *(const float4*)(cond + (size_t)row * Cn + c4 * 4);
        *(float4*)(x0 + (size_t)row * D0n + Pn + c4 * 4) = v;
    }
}

// ---------------------------------------------------------------------------
// xg = A[64000,512] @ W[2048,512].T + (b1 + b2)   -> Out[64000,2048]
// grid (Mrows/128, Gn/16), block 256; W tile staged in LDS, shared by 8 waves.
// ---------------------------------------------------------------------------
__global__ __launch_bounds__(256)
void k_gemm_xg(const float* __restrict__ A,  const float* __restrict__ W,
               const float* __restrict__ b1, const float* __restrict__ b2,
               float* __restrict__ Out) {
    __shared__ float sW[16 * (Hn + 4)];   // 16 x 516 floats = 33 KB

    const int tid   = threadIdx.x;
    const int n0    = blockIdx.y * 16;
    const int mBase = blockIdx.x * 128;

    for (int t4 = tid; t4 < 16 * (Hn / 4); t4 += 256) {
        int nl = t4 / (Hn / 4), kq = t4 % (Hn / 4);
        float4 v = *(const float4*)(W + (size_t)(n0 + nl) * Hn + kq * 4);
        *(float4*)(&sW[nl * (Hn + 4) + kq * 4]) = v;
    }
    __syncthreads();

    const int wave  = tid >> 5, lane = tid & 31;
    const int mrow  = lane & 15;
    const int khalf = lane >> 4;
    const int ncol  = lane & 15;
    const float* aRow = A + (size_t)(mBase + wave * 16 + mrow) * Hn + 2 * khalf;
    const float* bRow = &sW[ncol * (Hn + 4) + 2 * khalf];

    v8f c0 = {}, c1 = {}, c2 = {}, c3 = {};
    #pragma unroll 4
    for (int k0 = 0; k0 < Hn; k0 += 16) {
        c0 = wmma4(*(const v2f*)(aRow + k0),      *(const v2f*)(bRow + k0),      c0);
        c1 = wmma4(*(const v2f*)(aRow + k0 + 4),  *(const v2f*)(bRow + k0 + 4),  c1);
        c2 = wmma4(*(const v2f*)(aRow + k0 + 8),  *(const v2f*)(bRow + k0 + 8),  c2);
        c3 = wmma4(*(const v2f*)(aRow + k0 + 12), *(const v2f*)(bRow + k0 + 12), c3);
    }
    v8f acc = (c0 + c1) + (c2 + c3);

    const int n = n0 + ncol;
    const float bias = b1[n] + b2[n];
    #pragma unroll
    for (int r = 0; r < 8; ++r) {
        int mm = mBase + wave * 16 + r + 8 * khalf;
        Out[(size_t)mm * Gn + n] = acc[r] + bias;
    }
}

// ---------------------------------------------------------------------------
// Persistent LSTM recurrence. 32 blocks; block owns hidden units j0..j0+15
// (all 4 gates), whh slice (64 rows x 512) resident in LDS for all T steps.
// Cross-block step sync: per-step arrival counters (agent-scope acq/rel),
// h ping-pong double buffer. Cell state c stays in registers.
// ---------------------------------------------------------------------------
__global__ __launch_bounds__(256)
void k_lstm_rec(const float* __restrict__ whh, const float* __restrict__ xg,
                float* __restrict__ y, float* __restrict__ hb /* [2][Bn*Hn] */,
                int* __restrict__ cnt) {
    __shared__ float sW[64 * (Hn + 4)];   // 64 x 516 floats = 132,096 B
    __shared__ float stage[Bn * 64];      // [b][gate*16 + jl] = 8 KB

    const int tid = threadIdx.x;
    const int j0  = blockIdx.x * 16;

    // Stage whh rows {gate*512 + j0 + jl : gate 0..3, jl 0..15} once.
    for (int t4 = tid; t4 < 64 * (Hn / 4); t4 += 256) {
        int r = t4 / (Hn / 4), kq = t4 % (Hn / 4);
        int grow = ((r >> 4) * Hn) + j0 + (r & 15);
        float4 v = *(const float4*)(whh + (size_t)grow * Hn + kq * 4);
        *(float4*)(&sW[r * (Hn + 4) + kq * 4]) = v;
    }
    __syncthreads();

    const int wave  = tid >> 5, lane = tid & 31;
    const int gate  = wave & 3;          // i,f,g,o
    const int mtile = wave >> 2;         // batch halves 0..15 / 16..31
    const int mrow  = lane & 15;
    const int khalf = lane >> 4;
    const int ncol  = lane & 15;
    const float* bRow = &sW[(gate * 16 + ncol) * (Hn + 4) + 2 * khalf];

    float creg0 = 0.0f, creg1 = 0.0f;    // cell state, 2 (b,j) pairs per thread

    for (int t = 0; t < Tn; ++t) {
        const float* hprev = hb + (size_t)(t & 1) * (Bn * Hn);
        float*       hnext = hb + (size_t)((t + 1) & 1) * (Bn * Hn);

        if (t > 0) {
            if (tid == 0) {
                while (__hip_atomic_load(&cnt[t - 1], __ATOMIC_ACQUIRE,
                                         __HIP_MEMORY_SCOPE_AGENT) < NBLK_REC) {
                    __builtin_amdgcn_s_sleep(1);
                }
            }
            __syncthreads();
        }

        // 16x16x512 tile of gates = h_prev @ whh.T
        const float* aRow = hprev + (size_t)(mtile * 16 + mrow) * Hn + 2 * khalf;
        v8f c0 = {}, c1 = {}, c2 = {}, c3 = {};
        #pragma unroll 4
        for (int k0 = 0; k0 < Hn; k0 += 16) {
            c0 = wmma4(*(const v2f*)(aRow + k0),      *(const v2f*)(bRow + k0),      c0);
            c1 = wmma4(*(const v2f*)(aRow + k0 + 4),  *(const v2f*)(bRow + k0 + 4),  c1);
            c2 = wmma4(*(const v2f*)(aRow + k0 + 8),  *(const v2f*)(bRow + k0 + 8),  c2);
            c3 = wmma4(*(const v2f*)(aRow + k0 + 12), *(const v2f*)(bRow + k0 + 12), c3);
        }
        v8f acc = (c0 + c1) + (c2 + c3);

        // add precomputed input projection, stash to LDS for the cell update
        #pragma unroll
        for (int r = 0; r < 8; ++r) {
            int b = mtile * 16 + r + 8 * khalf;
            float xv = xg[((size_t)b * Tn + t) * Gn + (size_t)gate * Hn + j0 + ncol];
            stage[b * 64 + gate * 16 + ncol] = acc[r] + xv;
        }
        __syncthreads();

        // elementwise LSTM cell: 32 b x 16 j = 512 cells, 2 per thread
        #pragma unroll
        for (int r2 = 0; r2 < 2; ++r2) {
            int idx = tid + 256 * r2;
            int b = idx >> 4, jl = idx & 15;
            float gi = sigmoidf_fast(stage[b * 64 +      jl]);
            float gf = sigmoidf_fast(stage[b * 64 + 16 + jl]);
            float gg = tanhf(        stage[b * 64 + 32 + jl]);
            float go = sigmoidf_fast(stage[b * 64 + 48 + jl]);
            float c  = (r2 == 0) ? creg0 : creg1;
            c = gf * c + gi * gg;
            if (r2 == 0) creg0 = c; else creg1 = c;
            float h = go * tanhf(c);
            hnext[(size_t)b * Hn + j0 + jl]          = h;
            y[((size_t)b * Tn + t) * Hn + j0 + jl]   = h;
        }
        __threadfence();          // publish h stores device-wide
        __syncthreads();          // stage reads done; h stores fenced by all
        if (tid == 0) {
            __hip_atomic_fetch_add(&cnt[t], 1, __ATOMIC_RELEASE,
                                   __HIP_MEMORY_SCOPE_AGENT);
        }
    }
}

// ---------------------------------------------------------------------------
// FC head: out[m] = mask ? 0 : relu(dot(y1[m,:], fcw) + fcb); wave per row.
// ---------------------------------------------------------------------------
__global__ __launch_bounds__(256)
void k_fc(const float* __restrict__ y1, const float* __restrict__ fcw,
          const float* __restrict__ fcb, const uint8_t* __restrict__ masks,
          float* __restrict__ out) {
    const int wave = threadIdx.x >> 5, lane = threadIdx.x & 31;
    const int m = blockIdx.x * 8 + wave;
    float s = 0.0f;
    #pragma unroll 4
    for (int k = lane; k < Hn; k += 32)
        s = fmaf(y1[(size_t)m * Hn + k], fcw[k], s);
    #pragma unroll
    for (int off = 16; off > 0; off >>= 1) s += __shfl_down(s, off, 32);
    if (lane == 0) {
        float v = fmaxf(s + fcb[0], 0.0f);
        out[m] = masks[m] ? 0.0f : v;
    }
}

// ---------------------------------------------------------------------------
// host launcher
// ---------------------------------------------------------------------------
extern "C" void kernel_launch(void* const* d_in, const int* in_sizes, int n_in,
                              void* d_out, int out_size, void* d_ws, size_t ws_size,
                              hipStream_t stream) {
    const float*   inputs = (const float*)d_in[0];
    const float*   cond   = (const float*)d_in[1];
    const uint8_t* masks  = (const uint8_t*)d_in[2];
    const float*   pw0    = (const float*)d_in[3];
    const float*   pb0    = (const float*)d_in[4];
    const float*   pw1    = (const float*)d_in[5];
    const float*   pb1    = (const float*)d_in[6];
    const float*   wih0   = (const float*)d_in[7];
    const float*   whh0   = (const float*)d_in[8];
    const float*   bih0   = (const float*)d_in[9];
    const float*   bhh0   = (const float*)d_in[10];
    const float*   wih1   = (const float*)d_in[11];
    const float*   whh1   = (const float*)d_in[12];
    const float*   bih1   = (const float*)d_in[13];
    const float*   bhh1   = (const float*)d_in[14];
    const float*   fcw    = (const float*)d_in[15];
    const float*   fcb    = (const float*)d_in[16];
    float*         out    = (float*)d_out;

    // workspace layout (bytes)
    const size_t X0_B  = (size_t)Mrows * D0n * 4;       // 131,072,000 (also y1)
    const size_t XG_B  = (size_t)Mrows * Gn  * 4;       // 524,288,000
    const size_t Y0_B  = (size_t)Mrows * Hn  * 4;       // 131,072,000
    const size_t HB_B  = (size_t)2 * Bn * Hn * 4;       // 131,072 (ping-pong)
    char* base = (char*)d_ws;
    float* X0 = (float*)(base);
    float* XG = (float*)(base + X0_B);
    float* Y0 = (float*)(base + X0_B + XG_B);
    float* HB = (float*)(base + X0_B + XG_B + Y0_B);
    int*   CNT = (int*)  (base + X0_B + XG_B + Y0_B + HB_B);

    // 1. prenet + concat
    k_prenet<<<dim3(Mrows / 128, Pn / 16), 256, 0, stream>>>(inputs, pw0, pb0, pw1, pb1, X0);
    k_condcopy<<<(Mrows * (Cn / 4) + 255) / 256, 256, 0, stream>>>(cond, X0);

    // 2. layer 0: input projection, then persistent recurrence
    k_gemm_xg<<<dim3(Mrows / 128, Gn / 16), 256, 0, stream>>>(X0, wih0, bih0, bhh0, XG);
    k_init<<<8, 256, 0, stream>>>(CNT, HB);
    k_lstm_rec<<<NBLK_REC, 256, 0, stream>>>(whh0, XG, Y0, HB, CNT);

    // 3. layer 1
    k_gemm_xg<<<dim3(Mrows / 128, Gn / 16), 256, 0, stream>>>(Y0, wih1, bih1, bhh1, XG);
    k_init<<<8, 256, 0, stream>>>(CNT, HB);
    k_lstm_rec<<<NBLK_REC, 256, 0, stream>>>(whh1, XG, X0 /* y1 */, HB, CNT);

    // 4. FC head + mask
    k_fc<<<Mrows / 8, 256, 0, stream>>>(X0, fcw, fcb, masks, out);
}